// AttentionAwareCompressor_15195594293490
// MI455X (gfx1250) — compile-verified
//
#include <hip/hip_runtime.h>
#include <hip/hip_bf16.h>

// Problem constants (fixed by the reference setup_inputs)
constexpr int B  = 4;
constexpr int H  = 32;
constexpr int S  = 4096;
constexpr int D  = 128;
constexpr int Dh = 64;
constexpr int K2 = 2048;               // compressed length = S * 0.5
constexpr int TOK = B * H * S;         // 524288 tokens
constexpr int ROWS = B * H;            // 128 (B,H) rows

typedef float v2f __attribute__((ext_vector_type(2)));
typedef float v8f __attribute__((ext_vector_type(8)));

// ---------------------------------------------------------------------------
// Kernel 1: per-token magnitude importance + learned importance (MLP via WMMA)
// Block = 128 threads = 4 waves; each wave owns a 16-token M-tile.
// W1 is staged into LDS in a fragment-interleaved layout so each WMMA B
// fragment is a single contiguous ds_load_b64:
//   sW1[p*128 + 2n + (k&1)] = W1[k][n],  p = k>>1
// ---------------------------------------------------------------------------
__global__ __launch_bounds__(128) void importance_stage1(
    const float* __restrict__ key, const float* __restrict__ val,
    const float* __restrict__ W1,  const float* __restrict__ b1,
    const float* __restrict__ W2,  const float* __restrict__ b2,
    float* __restrict__ magOut, float* __restrict__ learnedOut)
{
    __shared__ float sComb[4 * 16 * D];   // 32 KB: combined rows per wave-tile
    __shared__ float sW1[D * Dh];         // 32 KB: W1, K-pair/column interleaved

    const int tid  = threadIdx.x;
    const int lane = tid & 31;
    const int wave = tid >> 5;

    // Stage W1 -> LDS interleaved. float4 over 4 consecutive n of one K row,
    // scattered as stride-2 b32 stores into the interleaved layout.
    const float4* w1v = (const float4*)W1;
    for (int i4 = tid; i4 < (D * Dh) / 4; i4 += 128) {
        const int k  = i4 >> 4;            // K row (Dh/4 = 16 float4 per row)
        const int n0 = (i4 & 15) * 4;      // first column of this float4
        const float4 v = w1v[i4];
        float* dst = &sW1[(k >> 1) * (2 * Dh) + 2 * n0 + (k & 1)];
        dst[0] = v.x; dst[2] = v.y; dst[4] = v.z; dst[6] = v.w;
    }

    const int tileBase = (blockIdx.x * 4 + wave) * 16;  // first token of tile
    float* myComb = &sComb[wave * 16 * D];

    // Stage combined = k + v into LDS; compute key/value magnitudes on the fly
    for (int j = 0; j < 16; ++j) {
        const int tok = tileBase + j;
        const float4 k4 = ((const float4*)(key + (size_t)tok * D))[lane];
        const float4 v4 = ((const float4*)(val + (size_t)tok * D))[lane];
        float4 c4;
        c4.x = k4.x + v4.x; c4.y = k4.y + v4.y;
        c4.z = k4.z + v4.z; c4.w = k4.w + v4.w;
        ((float4*)(myComb + j * D))[lane] = c4;
        float ssk = k4.x*k4.x + k4.y*k4.y + k4.z*k4.z + k4.w*k4.w;
        float ssv = v4.x*v4.x + v4.y*v4.y + v4.z*v4.z + v4.w*v4.w;
        #pragma unroll
        for (int off = 16; off > 0; off >>= 1) {
            ssk += __shfl_xor(ssk, off, 32);
            ssv += __shfl_xor(ssv, off, 32);
        }
        if (lane == 0)
            magOut[tok] = 0.5f * (sqrtf(ssk) + sqrtf(ssv));
    }
    __syncthreads();

    // ---- WMMA: H16x64 = Combined(16xD) @ W1(DxDh), fp32 16x16x4 tiles ----
    const int m    = lane & 15;        // M index (A) and N index (B/C)
    const int half = lane >> 4;        // 0: lanes 0-15, 1: lanes 16-31
    const int kr   = half * 2;         // K sub-offset for A and B fragments

    v8f acc[4];
    #pragma unroll
    for (int nt = 0; nt < 4; ++nt)
        acc[nt] = (v8f){0.f,0.f,0.f,0.f,0.f,0.f,0.f,0.f};

    for (int c = 0; c < 32; ++c) {            // K chunks of 4 (D = 128)
        const int kb = 4 * c + kr;            // even
        const int p  = kb >> 1;               // K-pair index = 2c + half
        // A fragment: 16x4 fp32 (element r -> K = kb + r)
        const float2 a2 = *(const float2*)&myComb[m * D + kb];
        v2f a; a.x = a2.x; a.y = a2.y;
        #pragma unroll
        for (int nt = 0; nt < 4; ++nt) {
            const int n = nt * 16 + m;
            // B fragment: contiguous (W1[kb][n], W1[kb+1][n]) -> one b64 load
            const float2 b2f = *(const float2*)&sW1[p * (2 * Dh) + 2 * n];
            v2f b; b.x = b2f.x; b.y = b2f.y;
            acc[nt] = __builtin_amdgcn_wmma_f32_16x16x4_f32(
                false, a, false, b, (short)0, acc[nt], false, false);
        }
    }

    // Epilogue: h = relu(acc + b1); learned = sigmoid(h . W2 + b2)
    // C/D layout: lane holds N = m; VGPR r holds token M = r + 8*half
    float partial[8];
    #pragma unroll
    for (int r = 0; r < 8; ++r) partial[r] = 0.f;
    #pragma unroll
    for (int nt = 0; nt < 4; ++nt) {
        const int f  = nt * 16 + m;
        const float w2 = W2[f];
        const float bb = b1[f];
        #pragma unroll
        for (int r = 0; r < 8; ++r) {
            float h = acc[nt][r] + bb;
            h = h > 0.f ? h : 0.f;
            partial[r] += h * w2;
        }
    }
    // Reduce across the 16 lanes of each half (same token set per half)
    #pragma unroll
    for (int off = 1; off < 16; off <<= 1)
        #pragma unroll
        for (int r = 0; r < 8; ++r)
            partial[r] += __shfl_xor(partial[r], off, 32);

    if (m == 0) {
        const float b2v = b2[0];
        #pragma unroll
        for (int r = 0; r < 8; ++r) {
            const int tok = tileBase + r + 8 * half;
            learnedOut[tok] = 1.f / (1.f + expf(-(partial[r] + b2v)));
        }
    }
}

// ---------------------------------------------------------------------------
// Kernel 2: per-(B,H)-row L2 normalization of magnitude + final importance
// ---------------------------------------------------------------------------
__global__ __launch_bounds__(256) void importance_stage2(
    const float* __restrict__ mag, const float* __restrict__ learned,
    float* __restrict__ importance)
{
    __shared__ float red[8];
    const int row = blockIdx.x;
    const float* m = mag + (size_t)row * S;

    float ss = 0.f;
    for (int i = threadIdx.x; i < S; i += 256) { float x = m[i]; ss += x * x; }
    #pragma unroll
    for (int off = 16; off > 0; off >>= 1) ss += __shfl_xor(ss, off, 32);
    if ((threadIdx.x & 31) == 0) red[threadIdx.x >> 5] = ss;
    __syncthreads();

    float tot = 0.f;
    #pragma unroll
    for (int w = 0; w < 8; ++w) tot += red[w];
    const float rinv = 1.f / fmaxf(sqrtf(tot), 1e-12f);
    const float attn = 0.2f * (1.0f / 64.0f);   // 0.2 / sqrt(4096)

    for (int i = threadIdx.x; i < S; i += 256) {
        const size_t o = (size_t)row * S + i;
        importance[o] = 0.4f * m[i] * rinv + 0.4f * learned[o] + attn;
    }
}

// ---------------------------------------------------------------------------
// Kernel 3: per-row descending bitonic sort (tie -> lowest index, like
// jax.lax.top_k); emit top-2048 indices (sorted) + compression mask.
// ---------------------------------------------------------------------------
__global__ __launch_bounds__(1024) void topk_sort(
    const float* __restrict__ importance, int* __restrict__ topkIdx,
    float* __restrict__ mask)
{
    __shared__ unsigned long long keys[S];   // 32 KB
    __shared__ float smask[S];               // 16 KB
    const int row = blockIdx.x;
    const int tid = threadIdx.x;

    for (int i = tid; i < S; i += 1024) {
        const float f = importance[(size_t)row * S + i];
        unsigned u = __float_as_uint(f);
        u = (u & 0x80000000u) ? ~u : (u | 0x80000000u);   // orderable float
        keys[i] = ((unsigned long long)u << 32) | (unsigned)(~(unsigned)i);
        smask[i] = 0.f;
    }
    __syncthreads();

    for (int k = 2; k <= S; k <<= 1) {
        for (int j = k >> 1; j > 0; j >>= 1) {
            for (int t = tid; t < S / 2; t += 1024) {
                const int i   = 2 * t - (t & (j - 1));
                const int ixj = i ^ j;
                const bool desc = ((i & k) == 0);
                const unsigned long long a = keys[i];
                const unsigned long long c = keys[ixj];
                if ((a < c) == desc) { keys[i] = c; keys[ixj] = a; }
            }
            __syncthreads();
        }
    }

    for (int kk = tid; kk < K2; kk += 1024) {
        const unsigned idx = ~(unsigned)(keys[kk] & 0xFFFFFFFFu);
        topkIdx[(size_t)row * K2 + kk] = (int)idx;
        smask[idx] = 1.f;
    }
    __syncthreads();
    for (int i = tid; i < S; i += 1024)
        mask[(size_t)row * S + i] = smask[i];
}

// ---------------------------------------------------------------------------
// Kernel 4: gather compressed K/V rows (bandwidth-dominant; float4 streams)
// ---------------------------------------------------------------------------
__global__ __launch_bounds__(64) void gather_kv(
    const float* __restrict__ key, const float* __restrict__ val,
    const int* __restrict__ topkIdx,
    float* __restrict__ outK, float* __restrict__ outV)
{
    const int g   = blockIdx.x;       // 0 .. ROWS*K2-1
    const int row = g / K2;
    const int idx = topkIdx[g];
    const size_t src = ((size_t)row * S + (size_t)idx) * D;
    const size_t dst = (size_t)g * D;
    const int lane = threadIdx.x & 31;
    if (threadIdx.x < 32)
        ((float4*)(outK + dst))[lane] = ((const float4*)(key + src))[lane];
    else
        ((float4*)(outV + dst))[lane] = ((const float4*)(val + src))[lane];
}

// ---------------------------------------------------------------------------
extern "C" void kernel_launch(void* const* d_in, const int* in_sizes, int n_in,
                              void* d_out, int out_size, void* d_ws, size_t ws_size,
                              hipStream_t stream) {
    const float* key = (const float*)d_in[0];
    const float* val = (const float*)d_in[1];
    const float* W1  = (const float*)d_in[2];
    const float* b1  = (const float*)d_in[3];
    const float* W2  = (const float*)d_in[4];
    const float* b2  = (const float*)d_in[5];

    // Workspace layout (all rewritten every call): ~7.3 MB
    float* ws         = (float*)d_ws;
    float* mag        = ws;                   // TOK
    float* learned    = ws + (size_t)TOK;     // TOK
    float* importance = ws + (size_t)2 * TOK; // TOK
    int*   topkIdx    = (int*)(ws + (size_t)3 * TOK); // ROWS*K2

    // Output layout: keys [ROWS*K2*D] | values [ROWS*K2*D] | mask [ROWS*S]
    float* outK = (float*)d_out;
    float* outV = outK + (size_t)ROWS * K2 * D;
    float* mask = outV + (size_t)ROWS * K2 * D;

    importance_stage1<<<TOK / 64, 128, 0, stream>>>(key, val, W1, b1, W2, b2,
                                                    mag, learned);
    importance_stage2<<<ROWS, 256, 0, stream>>>(mag, learned, importance);
    topk_sort<<<ROWS, 1024, 0, stream>>>(importance, topkIdx, mask);
    gather_kv<<<ROWS * K2, 64, 0, stream>>>(key, val, topkIdx, outK, outV);
}